// L_memory_dynamic_temperature_45234595561871
// MI455X (gfx1250) — compile-verified
//
#include <hip/hip_runtime.h>
#include <math.h>

typedef __attribute__((ext_vector_type(2))) float v2f;
typedef __attribute__((ext_vector_type(8))) float v8f;

#define N_ROWS      2048
#define M_COLS      32768
#define DDIM        128
#define BASE_TEMP   0.07f
#define EPSILON_T   0.01f

#define WAVES_PER_BLOCK 8
#define BLOCK_THREADS   (WAVES_PER_BLOCK * 32)
#define ROW_TILE        16
#define COL_TILE        16
#define NUM_ROW_BLOCKS  (N_ROWS / ROW_TILE)        // 128
#define NUM_COL_TILES   (M_COLS / COL_TILE)        // 2048
#define KSTEPS          (DDIM / 4)                 // 32 wmma_f32_16x16x4 / tile
#define COL_SPLITS      8                          // grid = 128 x 8 = 1024 blocks
#define TILES_PER_SPLIT (NUM_COL_TILES / COL_SPLITS) // 256 (32 per wave)

// d_ws layout:
//   stats  : float4[N_ROWS * COL_SPLITS]  (max, sumexp, S1, n_pos)   256 KB
//   rowLoss: float [N_ROWS]               at offset N_ROWS*COL_SPLITS*16B
#define STATS_F4_COUNT (N_ROWS * COL_SPLITS)

__global__ __launch_bounds__(BLOCK_THREADS)
void fused_sim_loss_kernel(const float* __restrict__ featOn,   // [N,D]
                           const int*   __restrict__ yOn,      // [N]
                           const float* __restrict__ featOff,  // [M,D]
                           const int*   __restrict__ yOff,     // [M]
                           const float* __restrict__ confOn,   // [N]
                           float4*      __restrict__ stats)    // [N_ROWS*COL_SPLITS]
{
    __shared__ float ldsMax[WAVES_PER_BLOCK][ROW_TILE];
    __shared__ float ldsSum[WAVES_PER_BLOCK][ROW_TILE];
    __shared__ float ldsS1 [WAVES_PER_BLOCK][ROW_TILE];
    __shared__ float ldsNp [WAVES_PER_BLOCK][ROW_TILE];

    const int rowBase = blockIdx.x * ROW_TILE;
    const int split   = blockIdx.y;
    const int wave    = threadIdx.x >> 5;
    const int lane    = threadIdx.x & 31;
    const int m16     = lane & 15;
    const int half    = lane >> 4;

    // ---- Hoist A fragments (tile-invariant): 32-bit A 16x4 layout ----
    // lane l holds row (l&15); reg pair holds K = 4*kk + 2*half + {0,1}
    v2f aF[KSTEPS];
    {
        const float* ap = featOn + (size_t)(rowBase + m16) * DDIM + 2 * half;
        #pragma unroll
        for (int kk = 0; kk < KSTEPS; ++kk)
            aF[kk] = *(const v2f*)(ap + kk * 4);
    }

    // Per-lane row scalars: acc element r maps to row rowBase + r + 8*half
    int   yRow[8];
    float cRow[8];
    #pragma unroll
    for (int r = 0; r < 8; ++r) {
        const int row = rowBase + r + 8 * half;
        yRow[r] = yOn[row];
        cRow[r] = confOn[row];
    }

    // Per-LANE online softmax state (no cross-lane traffic inside the loop)
    float runMax[8], runSum[8], s1[8], np[8];
    #pragma unroll
    for (int r = 0; r < 8; ++r) {
        runMax[r] = -3.0e38f;
        runSum[r] = 0.0f;
        s1[r]     = 0.0f;
        np[r]     = 0.0f;
    }

    const int tileBegin = split * TILES_PER_SPLIT;
    const int tileEnd   = tileBegin + TILES_PER_SPLIT;

    for (int tile = tileBegin + wave; tile < tileEnd; tile += WAVES_PER_BLOCK) {
        const int col = tile * COL_TILE + m16;
        const int yc  = yOff[col];
        const float* bp = featOff + (size_t)col * DDIM + 2 * half;

        // Bulk-stage all B fragments -> independent loads, deep pipeline
        v2f bF[KSTEPS];
        #pragma unroll
        for (int kk = 0; kk < KSTEPS; ++kk)
            bF[kk] = *(const v2f*)(bp + kk * 4);

        // Two independent accumulator chains -> overlap WMMA RAW latency
        v8f acc0 = {0.0f, 0.0f, 0.0f, 0.0f, 0.0f, 0.0f, 0.0f, 0.0f};
        v8f acc1 = {0.0f, 0.0f, 0.0f, 0.0f, 0.0f, 0.0f, 0.0f, 0.0f};
        #pragma unroll
        for (int kk = 0; kk < KSTEPS / 2; ++kk) {
            acc0 = __builtin_amdgcn_wmma_f32_16x16x4_f32(
                false, aF[kk], false, bF[kk], (short)0, acc0, false, false);
            acc1 = __builtin_amdgcn_wmma_f32_16x16x4_f32(
                false, aF[kk + KSTEPS / 2], false, bF[kk + KSTEPS / 2],
                (short)0, acc1, false, false);
        }

        // ---- Fused epilogue: dynamic temperature + per-lane online softmax ----
        #pragma unroll
        for (int r = 0; r < 8; ++r) {
            const float simr     = acc0[r] + acc1[r];
            const bool  pos      = (yRow[r] == yc);
            const float featConf = pos ? simr : (1.0f - simr);
            const float combined = 0.7f * featConf + 0.3f * cRow[r];
            const float sig      = __builtin_amdgcn_rcpf(1.0f + __expf(-2.0f * combined));
            const float confW    = 0.1f + 0.9f * sig;
            const float simsc    = simr * ((confW + EPSILON_T) * (1.0f / BASE_TEMP));

            s1[r] += pos ? simsc : 0.0f;
            np[r] += pos ? 1.0f  : 0.0f;

            // one-exp online (max, sumexp) update, branchless
            const float rm = runMax[r];
            const float d  = __expf(-fabsf(simsc - rm));
            const bool  gt = (simsc > rm);
            runSum[r] = gt ? (runSum[r] * d + 1.0f) : (runSum[r] + d);
            runMax[r] = gt ? simsc : rm;
        }
    }

    // ---- Merge (max,sumexp) and sum S1/n_pos across the 16-lane half ----
    #pragma unroll
    for (int r = 0; r < 8; ++r) {
        float m = runMax[r];
        float s = runSum[r];
        #pragma unroll
        for (int mask = 1; mask <= 8; mask <<= 1) {
            const float om = __shfl_xor(m, mask, 32);
            const float os = __shfl_xor(s, mask, 32);
            const float nm = fmaxf(m, om);
            s = s * __expf(m - nm) + os * __expf(om - nm);
            m = nm;
        }
        runMax[r] = m;
        runSum[r] = s;

        float v = s1[r];
        v += __shfl_xor(v, 1, 32);
        v += __shfl_xor(v, 2, 32);
        v += __shfl_xor(v, 4, 32);
        v += __shfl_xor(v, 8, 32);
        s1[r] = v;

        float n = np[r];
        n += __shfl_xor(n, 1, 32);
        n += __shfl_xor(n, 2, 32);
        n += __shfl_xor(n, 4, 32);
        n += __shfl_xor(n, 8, 32);
        np[r] = n;
    }

    if (m16 == 0) {
        #pragma unroll
        for (int r = 0; r < 8; ++r) {
            const int row = r + 8 * half;
            ldsMax[wave][row] = runMax[r];
            ldsSum[wave][row] = runSum[r];
            ldsS1 [wave][row] = s1[r];
            ldsNp [wave][row] = np[r];
        }
    }
    __syncthreads();

    // ---- Combine waves (fixed order), publish per-(row,split) stats ----
    if (threadIdx.x < ROW_TILE) {
        const int t = threadIdx.x;
        float m = -3.0e38f;
        for (int w = 0; w < WAVES_PER_BLOCK; ++w) m = fmaxf(m, ldsMax[w][t]);
        float sum = 0.0f, S1 = 0.0f, NP = 0.0f;
        for (int w = 0; w < WAVES_PER_BLOCK; ++w) {
            sum += ldsSum[w][t] * __expf(ldsMax[w][t] - m);
            S1  += ldsS1[w][t];
            NP  += ldsNp[w][t];
        }
        float4 st;
        st.x = m; st.y = sum; st.z = S1; st.w = NP;
        stats[(size_t)(rowBase + t) * COL_SPLITS + split] = st;
    }
}

// ---- Stage 2: merge column splits per row (fixed order), emit per-row loss ----
__global__ __launch_bounds__(256)
void combine_rows_kernel(const float4* __restrict__ stats,
                         const float*  __restrict__ confOn,
                         float*        __restrict__ rowLoss)
{
    const int row = blockIdx.x * 256 + threadIdx.x;
    if (row >= N_ROWS) return;

    float m = -3.0e38f;
    #pragma unroll
    for (int s = 0; s < COL_SPLITS; ++s)
        m = fmaxf(m, stats[(size_t)row * COL_SPLITS + s].x);

    float sum = 0.0f, S1 = 0.0f, NP = 0.0f;
    #pragma unroll
    for (int s = 0; s < COL_SPLITS; ++s) {
        const float4 st = stats[(size_t)row * COL_SPLITS + s];
        sum += st.y * __expf(st.x - m);
        S1  += st.z;
        NP  += st.w;
    }

    const float logZ  = logf(sum + 1e-8f);
    const float npEff = (NP < 1e-6f) ? 1.0f : NP;
    const float conf  = confOn[row];
    // Sum_j conf*mask*log_prob = conf*(S1 - NP*(max + logZ)); loss = -that/npEff
    rowLoss[row] = -(conf * (S1 - NP * (m + logZ))) / npEff;
}

// ---- Stage 3: deterministic two-level reduction of 2048 row losses ----
__global__ __launch_bounds__(256)
void finalize_kernel(const float* __restrict__ rowLoss,
                     float* __restrict__ out)
{
    __shared__ float lds[256];
    const int t = threadIdx.x;
    float s = 0.0f;
    #pragma unroll
    for (int i = 0; i < N_ROWS / 256; ++i)   // fixed-order per-thread partial
        s += rowLoss[t + i * 256];
    lds[t] = s;
    __syncthreads();
    if (t == 0) {
        float tot = 0.0f;
        for (int i = 0; i < 256; ++i) tot += lds[i];   // fixed order
        out[0] = tot / (float)N_ROWS;
    }
}

extern "C" void kernel_launch(void* const* d_in, const int* in_sizes, int n_in,
                              void* d_out, int out_size, void* d_ws, size_t ws_size,
                              hipStream_t stream) {
    const float* featOn  = (const float*)d_in[0];   // [2048,128]  f32
    const int*   yOn     = (const int*)  d_in[1];   // [2048]      i32
    const float* featOff = (const float*)d_in[2];   // [32768,128] f32
    const int*   yOff    = (const int*)  d_in[3];   // [32768]     i32
    const float* confOn  = (const float*)d_in[4];   // [2048]      f32
    float*       out     = (float*)d_out;           // scalar

    float4* stats   = (float4*)d_ws;                             // 256 KB
    float*  rowLoss = (float*)((char*)d_ws + STATS_F4_COUNT * sizeof(float4)); // 8 KB

    dim3 grid(NUM_ROW_BLOCKS, COL_SPLITS);
    fused_sim_loss_kernel<<<grid, BLOCK_THREADS, 0, stream>>>(
        featOn, yOn, featOff, yOff, confOn, stats);
    combine_rows_kernel<<<N_ROWS / 256, 256, 0, stream>>>(stats, confOn, rowLoss);
    finalize_kernel<<<1, 256, 0, stream>>>(rowLoss, out);
}